// AffinityDiffProposedModel_86431921865308
// MI455X (gfx1250) — compile-verified
//
#include <hip/hip_runtime.h>
#include <math.h>

// ---------------------------------------------------------------------------
// Problem constants (from reference): B=4, L=256, D=512, H=8, DH=64
// ---------------------------------------------------------------------------
#define BATCH 4
#define SEQ   256
#define DMODEL 512
#define HEADS 8
#define DHEAD 64
#define ROWS  (BATCH * SEQ)          // 1024 rows for (B*L, D) views

typedef float v2f __attribute__((ext_vector_type(2)));
typedef float v8f __attribute__((ext_vector_type(8)));

// V_WMMA_F32_16X16X4_F32: D(16x16,f32) = A(16x4,f32) * B(4x16,f32) + C
// 8-arg form: (neg_a, A, neg_b, B, c_mod, C, reuse_a, reuse_b)
static __device__ __forceinline__ v8f wmma4(v2f a, v2f b, v8f c) {
    return __builtin_amdgcn_wmma_f32_16x16x4_f32(
        false, a, false, b, (short)0, c, false, false);
}

// ---------------------------------------------------------------------------
// 32x32-per-wave GEMM tile:  P = act(A(1024x512-view) @ W(512x512) + bias)
// 4 independent accumulators -> 4 WMMAs per k-step reusing each operand 2x
// (8 FLOP/byte from L2 instead of 4 with a single 16x16 tile).
// Operand layout per CDNA5 ISA 7.12.2 (f32 16x16x4, wave32).
// ---------------------------------------------------------------------------
static __device__ __forceinline__ void gemm512_tile32(
    const float* __restrict__ A, const float* __restrict__ Wm,
    const float* __restrict__ bias, float* __restrict__ P,
    int m0, int n0, int lane, bool relu)
{
    const int l16 = lane & 15;
    const int kh  = lane >> 4;      // which K half-pair this lane holds
    v8f acc00 = {}, acc01 = {}, acc10 = {}, acc11 = {};
    const float* arow0 = A + (size_t)(m0 + l16) * DMODEL + 2 * kh;
    const float* arow1 = arow0 + 16 * DMODEL;
    const float* wcol  = Wm + (size_t)(2 * kh) * DMODEL + n0 + l16;
    #pragma unroll 4
    for (int k0 = 0; k0 < DMODEL; k0 += 4) {
        v2f a0 = *(const v2f*)(arow0 + k0);
        v2f a1 = *(const v2f*)(arow1 + k0);
        const float* w = wcol + (size_t)k0 * DMODEL;
        v2f b0, b1;
        b0.x = w[0];           b0.y = w[DMODEL];
        b1.x = w[16];          b1.y = w[DMODEL + 16];
        acc00 = wmma4(a0, b0, acc00);
        acc01 = wmma4(a0, b1, acc01);
        acc10 = wmma4(a1, b0, acc10);
        acc11 = wmma4(a1, b1, acc11);
    }
    const float bv0 = bias ? bias[n0 + l16]      : 0.0f;
    const float bv1 = bias ? bias[n0 + 16 + l16] : 0.0f;
    #pragma unroll
    for (int r = 0; r < 8; ++r) {
        const int row0 = m0 + r + 8 * kh;
        const int row1 = row0 + 16;
        const int col0 = n0 + l16;
        float v00 = acc00[r] + bv0, v01 = acc01[r] + bv1;
        float v10 = acc10[r] + bv0, v11 = acc11[r] + bv1;
        if (relu) {
            v00 = fmaxf(v00, 0.0f); v01 = fmaxf(v01, 0.0f);
            v10 = fmaxf(v10, 0.0f); v11 = fmaxf(v11, 0.0f);
        }
        P[(size_t)row0 * DMODEL + col0]      = v00;
        P[(size_t)row0 * DMODEL + col0 + 16] = v01;
        P[(size_t)row1 * DMODEL + col0]      = v10;
        P[(size_t)row1 * DMODEL + col0 + 16] = v11;
    }
}

// ---------------------------------------------------------------------------
// Kernel 1: Q/K/V projections  P = relu(X @ W + b), X=(1024,512)
// 32x32 tiles: (1024/32)*(512/32)=512 per matrix, x3 = 1536 waves = 192 blocks
// ---------------------------------------------------------------------------
__global__ __launch_bounds__(256)
void k_qkv_proj(const float* __restrict__ q, const float* __restrict__ k,
                const float* __restrict__ v,
                const float* __restrict__ Wq, const float* __restrict__ bq,
                const float* __restrict__ Wk, const float* __restrict__ bk,
                const float* __restrict__ Wv, const float* __restrict__ bv,
                float* __restrict__ Qp, float* __restrict__ Kp,
                float* __restrict__ Vp)
{
    const int lane = threadIdx.x & 31;
    const int gw   = blockIdx.x * 8 + (threadIdx.x >> 5);
    const int which = gw >> 9;             // / 512
    const int rem   = gw & 511;
    const int m0 = (rem >> 4) << 5;        // 32 M tiles of 32
    const int n0 = (rem & 15) << 5;        // 16 N tiles of 32

    const float* X; const float* W; const float* bias; float* P;
    if (which == 0)      { X = q; W = Wq; bias = bq; P = Qp; }
    else if (which == 1) { X = k; W = Wk; bias = bk; P = Kp; }
    else                 { X = v; W = Wv; bias = bv; P = Vp; }
    gemm512_tile32(X, W, bias, P, m0, n0, lane, /*relu=*/true);
}

// ---------------------------------------------------------------------------
// Kernel 2: fused euclidean diff-attention for one (head,batch) x 32-q-chunk.
//   S = Qc @ Kc^T (WMMA, 32x32 per wave) -> logits = -sqrt(qn+kn-2S)/8
//   -> softmax*mask -> ctx = P @ Vc (WMMA), merged-head write to ctx buffer
// grid = 32 headbatch * 8 qchunks = 256 blocks, 256 threads (8 waves)
// ---------------------------------------------------------------------------
__global__ __launch_bounds__(256)
void k_attention(const float* __restrict__ rawq,
                 const float* __restrict__ Qp, const float* __restrict__ Kp,
                 const float* __restrict__ Vp, float* __restrict__ ctxg)
{
    __shared__ float S[32][SEQ];      // 32 KB: scores then probabilities
    __shared__ float knorm[SEQ];
    __shared__ float qnorm[32];
    __shared__ float qmask[32];
    __shared__ float red[32][8];
    __shared__ float rowmax[32];
    __shared__ float rowsum[32];

    const int t    = threadIdx.x;
    const int lane = t & 31;
    const int wid  = t >> 5;
    const int hb   = blockIdx.x >> 3;     // 0..31 ; hb = h*B + b
    const int qc   = blockIdx.x & 7;      // 32-row query chunk
    const int h    = hb >> 2;             // / BATCH
    const int bb   = hb & 3;
    const int hoff = h * DHEAD;
    const size_t kbase = (size_t)(bb * SEQ) * DMODEL + hoff;           // K/V head slice
    const size_t qbase = (size_t)(bb * SEQ + qc * 32) * DMODEL + hoff;

    // --- key norms: one thread per key row ---
    {
        const float* kr = Kp + kbase + (size_t)t * DMODEL;
        float s = 0.0f;
        #pragma unroll 8
        for (int d = 0; d < DHEAD; ++d) { float x = kr[d]; s += x * x; }
        knorm[t] = s;
    }
    // --- query norms: threads 0..31 ---
    if (t < 32) {
        const float* qr = Qp + qbase + (size_t)t * DMODEL;
        float s = 0.0f;
        #pragma unroll 8
        for (int d = 0; d < DHEAD; ++d) { float x = qr[d]; s += x * x; }
        qnorm[t] = s;
    }
    // --- query mask from RAW queries: sign(|sum_d q|), 8 threads per row ---
    {
        const int row = t >> 3, seg = t & 7;
        const float* qr = rawq + (size_t)(bb * SEQ + qc * 32 + row) * DMODEL + seg * 64;
        float s = 0.0f;
        #pragma unroll 8
        for (int d = 0; d < 64; ++d) s += qr[d];
        red[row][seg] = s;
    }
    __syncthreads();
    if (t < 32) {
        float s = 0.0f;
        #pragma unroll
        for (int i = 0; i < 8; ++i) s += red[t][i];
        qmask[t] = (s != 0.0f) ? 1.0f : 0.0f;
    }
    __syncthreads();

    // --- S = Qc(32xDH) @ Kc^T(DHx256): one 32x32 region per wave ---
    const int l16 = lane & 15, kh = lane >> 4;
    {
        const int n0 = wid << 5;          // 8 waves x 32 keys
        v8f acc00 = {}, acc01 = {}, acc10 = {}, acc11 = {};
        const float* aq0 = Qp + qbase + (size_t)l16 * DMODEL + 2 * kh;
        const float* aq1 = aq0 + 16 * DMODEL;
        const float* bk0 = Kp + kbase + (size_t)(n0 + l16) * DMODEL + 2 * kh;
        const float* bk1 = bk0 + 16 * DMODEL;
        #pragma unroll
        for (int k0 = 0; k0 < DHEAD; k0 += 4) {
            v2f a0 = *(const v2f*)(aq0 + k0);
            v2f a1 = *(const v2f*)(aq1 + k0);
            v2f b0 = *(const v2f*)(bk0 + k0);   // K^T: contiguous in d
            v2f b1 = *(const v2f*)(bk1 + k0);
            acc00 = wmma4(a0, b0, acc00);
            acc01 = wmma4(a0, b1, acc01);
            acc10 = wmma4(a1, b0, acc10);
            acc11 = wmma4(a1, b1, acc11);
        }
        #pragma unroll
        for (int r = 0; r < 8; ++r) {
            const int i0 = r + 8 * kh, i1 = i0 + 16;
            const int j0 = n0 + l16,   j1 = j0 + 16;
            float d2;
            d2 = fmaxf(qnorm[i0] + knorm[j0] - 2.0f * acc00[r], 0.0f);
            S[i0][j0] = -sqrtf(d2) * 0.125f;     // / sqrt(DH)=8
            d2 = fmaxf(qnorm[i0] + knorm[j1] - 2.0f * acc01[r], 0.0f);
            S[i0][j1] = -sqrtf(d2) * 0.125f;
            d2 = fmaxf(qnorm[i1] + knorm[j0] - 2.0f * acc10[r], 0.0f);
            S[i1][j0] = -sqrtf(d2) * 0.125f;
            d2 = fmaxf(qnorm[i1] + knorm[j1] - 2.0f * acc11[r], 0.0f);
            S[i1][j1] = -sqrtf(d2) * 0.125f;
        }
    }
    __syncthreads();

    // --- softmax over keys, 8 threads per row ---
    const int srow = t >> 3, sseg = t & 7, c0 = sseg * 32;
    {
        float m = -INFINITY;
        for (int j = 0; j < 32; ++j) m = fmaxf(m, S[srow][c0 + j]);
        red[srow][sseg] = m;
    }
    __syncthreads();
    if (t < 32) {
        float m = -INFINITY;
        #pragma unroll
        for (int i = 0; i < 8; ++i) m = fmaxf(m, red[t][i]);
        rowmax[t] = m;
    }
    __syncthreads();
    {
        const float m = rowmax[srow];
        float s = 0.0f;
        for (int j = 0; j < 32; ++j) {
            float e = __expf(S[srow][c0 + j] - m);
            S[srow][c0 + j] = e;
            s += e;
        }
        red[srow][sseg] = s;
    }
    __syncthreads();
    if (t < 32) {
        float s = 0.0f;
        #pragma unroll
        for (int i = 0; i < 8; ++i) s += red[t][i];
        rowsum[t] = s;
    }
    __syncthreads();
    {
        const float scale = qmask[srow] / rowsum[srow];
        for (int j = 0; j < 32; ++j) S[srow][c0 + j] *= scale;
    }
    __syncthreads();

    // --- ctx = P(32x256) @ V(256x64): 2x4 16-tiles, 1 per wave ---
    {
        const int m0 = (wid >> 2) << 4;      // 0 or 16
        const int n0 = (wid & 3) << 4;       // 0..48 within head dim
        v8f acc = {};
        const float* bvp = Vp + kbase + n0 + l16 + (size_t)(2 * kh) * DMODEL;
        #pragma unroll 4
        for (int k0 = 0; k0 < SEQ; k0 += 4) {
            v2f a = *(const v2f*)(&S[m0 + l16][k0 + 2 * kh]);
            v2f b;
            b.x = bvp[(size_t)k0 * DMODEL];
            b.y = bvp[(size_t)k0 * DMODEL + DMODEL];
            acc = wmma4(a, b, acc);
        }
        #pragma unroll
        for (int r = 0; r < 8; ++r) {
            const int row = m0 + r + 8 * kh;
            const int col = n0 + l16;
            ctxg[(size_t)(bb * SEQ + qc * 32 + row) * DMODEL + hoff + col] = acc[r];
        }
    }
}

// ---------------------------------------------------------------------------
// Kernel 3: inter = ctx @ W_bil  (1024x512 @ 512x512), no bias/act
// 32x32 tiles = 512 waves = 64 blocks
// ---------------------------------------------------------------------------
__global__ __launch_bounds__(256)
void k_bil_inter(const float* __restrict__ ctxg, const float* __restrict__ Wbil,
                 float* __restrict__ inter)
{
    const int lane = threadIdx.x & 31;
    const int gw   = blockIdx.x * 8 + (threadIdx.x >> 5);
    const int m0 = (gw >> 4) << 5;
    const int n0 = (gw & 15) << 5;
    gemm512_tile32(ctxg, Wbil, nullptr, inter, m0, n0, lane, /*relu=*/false);
}

// ---------------------------------------------------------------------------
// Kernel 4: scores[b] = inter[b] @ keys[b]^T + bias  -> (4,256,256)
// 32x32 tiles: 8x8 per batch, x4 = 256 waves = 32 blocks
// Both operands are row-contiguous (keys^T pattern) -> all-b64 loads.
// ---------------------------------------------------------------------------
__global__ __launch_bounds__(256)
void k_bil_scores(const float* __restrict__ inter, const float* __restrict__ keys,
                  const float* __restrict__ bbil, float* __restrict__ out)
{
    const int lane = threadIdx.x & 31;
    const int l16 = lane & 15, kh = lane >> 4;
    const int gw  = blockIdx.x * 8 + (threadIdx.x >> 5);
    const int bb  = gw >> 6;
    const int rem = gw & 63;
    const int m0 = (rem >> 3) << 5;
    const int n0 = (rem & 7) << 5;

    const float* arow0 = inter + (size_t)(bb * SEQ + m0 + l16) * DMODEL + 2 * kh;
    const float* arow1 = arow0 + 16 * DMODEL;
    const float* krow0 = keys  + (size_t)(bb * SEQ + n0 + l16) * DMODEL + 2 * kh;
    const float* krow1 = krow0 + 16 * DMODEL;
    v8f acc00 = {}, acc01 = {}, acc10 = {}, acc11 = {};
    #pragma unroll 4
    for (int k0 = 0; k0 < DMODEL; k0 += 4) {
        v2f a0 = *(const v2f*)(arow0 + k0);
        v2f a1 = *(const v2f*)(arow1 + k0);
        v2f b0 = *(const v2f*)(krow0 + k0);   // keys^T: contiguous along e
        v2f b1 = *(const v2f*)(krow1 + k0);
        acc00 = wmma4(a0, b0, acc00);
        acc01 = wmma4(a0, b1, acc01);
        acc10 = wmma4(a1, b0, acc10);
        acc11 = wmma4(a1, b1, acc11);
    }
    const float bias = bbil[0];
    float* ob = out + (size_t)bb * SEQ * SEQ;
    #pragma unroll
    for (int r = 0; r < 8; ++r) {
        const int row0 = m0 + r + 8 * kh, row1 = row0 + 16;
        const int col0 = n0 + l16,        col1 = col0 + 16;
        ob[(size_t)row0 * SEQ + col0] = acc00[r] + bias;
        ob[(size_t)row0 * SEQ + col1] = acc01[r] + bias;
        ob[(size_t)row1 * SEQ + col0] = acc10[r] + bias;
        ob[(size_t)row1 * SEQ + col1] = acc11[r] + bias;
    }
}

// ---------------------------------------------------------------------------
// Host entry
// ---------------------------------------------------------------------------
extern "C" void kernel_launch(void* const* d_in, const int* in_sizes, int n_in,
                              void* d_out, int out_size, void* d_ws, size_t ws_size,
                              hipStream_t stream) {
    const float* q    = (const float*)d_in[0];
    const float* k    = (const float*)d_in[1];
    const float* v    = (const float*)d_in[2];
    const float* Wq   = (const float*)d_in[3];
    const float* bq   = (const float*)d_in[4];
    const float* Wk   = (const float*)d_in[5];
    const float* bk   = (const float*)d_in[6];
    const float* Wv   = (const float*)d_in[7];
    const float* bv   = (const float*)d_in[8];
    const float* Wbil = (const float*)d_in[9];
    const float* bbil = (const float*)d_in[10];
    float* out = (float*)d_out;

    float* ws = (float*)d_ws;
    const size_t MAT = (size_t)ROWS * DMODEL;   // 524288 floats
    float* Qp    = ws + 0 * MAT;
    float* Kp    = ws + 1 * MAT;
    float* Vp    = ws + 2 * MAT;
    float* ctxg  = ws + 3 * MAT;
    float* inter = ws + 4 * MAT;                // total 10.5 MB

    k_qkv_proj<<<192, 256, 0, stream>>>(q, k, v, Wq, bq, Wk, bk, Wv, bv,
                                        Qp, Kp, Vp);
    k_attention<<<256, 256, 0, stream>>>(q, Qp, Kp, Vp, ctxg);
    k_bil_inter<<<64, 256, 0, stream>>>(ctxg, Wbil, inter);
    k_bil_scores<<<32, 256, 0, stream>>>(inter, k, bbil, out);
}